// FinancialForecastingModel_75273596830207
// MI455X (gfx1250) — compile-verified
//
#include <hip/hip_runtime.h>
#include <hip/hip_bf16.h>
#include <cstdint>

// Problem sizes (fixed by the reference)
#define N_NODES   50000
#define N_EDGES   800000
#define F_DIM     128      // F_IN == HID == 128
#define N_CLASSES 10

typedef __attribute__((ext_vector_type(2))) float v2f;
typedef __attribute__((ext_vector_type(8))) float v8f;

// LDS row stride for the weight tile: 132 floats -> bank(n*132+k) = (4n+k)%64,
// conflict-free across lanes 0-15 (distinct 4n) and disjoint from lanes 16-31
// (k parity differs by 2 mod 4). Also keeps float2 loads 8B-aligned (k even).
#define WLDS_STRIDE 132

// ---------------------------------------------------------------------------
// Zero-fill (agg buffers must start at 0; harness poisons d_ws).
// ---------------------------------------------------------------------------
__global__ void zero_f32(float* __restrict__ p, int n) {
  int i = blockIdx.x * blockDim.x + threadIdx.x;
  if (i < n) p[i] = 0.0f;
}

// ---------------------------------------------------------------------------
// Edge scatter-add: agg[dst] += x[src].  One wave per edge; lane f covers
// features f, f+32, f+64, f+96 (fully coalesced 128B gathers, then
// global_atomic_add_f32 into the L2-resident agg buffer).
// ---------------------------------------------------------------------------
__global__ void scatter_add_128(const float* __restrict__ x,
                                const int* __restrict__ src,
                                const int* __restrict__ dst,
                                float* __restrict__ agg, int n_edges) {
  int gid  = blockIdx.x * blockDim.x + threadIdx.x;
  int e    = gid >> 5;
  int lane = gid & 31;
  if (e >= n_edges) return;
  int s = src[e];
  int d = dst[e];
  const float* xs = x   + (size_t)s * F_DIM;
  float*       ad = agg + (size_t)d * F_DIM;
#pragma unroll
  for (int j = 0; j < 4; ++j)
    atomicAdd(&ad[lane + 32 * j], xs[lane + 32 * j]);
}

// ---------------------------------------------------------------------------
// WMMA fp32 GEMM:  out = act( A0 @ W0^T [+ A1 @ W1^T] + bias )
//   A*: [n_rows x 128] row-major,  W*: [n_cols x 128] row-major (so W^T is B)
//   out: [n_rows x n_cols] row-major.
// One wave -> 16 rows x (16*NCOL_TILES) cols using V_WMMA_F32_16X16X4_F32,
// K=128 in 32 steps of 4.  W staged in padded LDS, reused by all waves.
// ---------------------------------------------------------------------------
template <int NCOL_TILES, bool RELU, bool DUAL>
__global__ void gemm_wmma_kernel(const float* __restrict__ A0,
                                 const float* __restrict__ W0,
                                 const float* __restrict__ A1,
                                 const float* __restrict__ W1,
                                 const float* __restrict__ bias,
                                 float* __restrict__ out,
                                 int n_rows, int n_cols) {
  extern __shared__ float smw[];  // [NCOL_TILES*16][WLDS_STRIDE]

  const int lane = threadIdx.x & 31;
  const int wave = threadIdx.x >> 5;
  const int row0 = (blockIdx.x * (blockDim.x >> 5) + wave) * 16;

  const int m    = lane & 15;  // row within A-tile / col within B-tile
  const int hi   = lane >> 4;  // half-wave selector
  const int krem = 2 * hi;     // K offset inside each K=4 group

  v8f c[NCOL_TILES];
#pragma unroll
  for (int t = 0; t < NCOL_TILES; ++t)
#pragma unroll
    for (int i = 0; i < 8; ++i) c[t][i] = 0.0f;

  // Clamp the A row for out-of-range tail waves (stores are guarded below;
  // no branch -> EXEC stays all-ones for the WMMAs).
  int ar = row0 + m;
  if (ar > n_rows - 1) ar = n_rows - 1;

  // ---- stage W0 into LDS (rows >= n_cols zero-filled for the fc2 case) ----
  for (int idx = threadIdx.x; idx < NCOL_TILES * 16 * F_DIM; idx += blockDim.x) {
    int r = idx >> 7, cc = idx & 127;
    smw[r * WLDS_STRIDE + cc] = (r < n_cols) ? W0[r * F_DIM + cc] : 0.0f;
  }
  __syncthreads();

  // ---- term 0: A0 @ W0^T ----
  {
    const float* __restrict__ arow = A0 + (size_t)ar * F_DIM;
    for (int k0 = 0; k0 < F_DIM; k0 += 4) {
      const int kk = k0 + krem;
      v2f a = *(const v2f*)(arow + kk);
#pragma unroll
      for (int t = 0; t < NCOL_TILES; ++t) {
        v2f b = *(const v2f*)(smw + (t * 16 + m) * WLDS_STRIDE + kk);
        c[t] = __builtin_amdgcn_wmma_f32_16x16x4_f32(
            false, a, false, b, (short)0, c[t], false, false);
      }
    }
  }

  // ---- term 1 (optional): A1 @ W1^T, reusing the same LDS buffer ----
  if (DUAL) {
    __syncthreads();
    for (int idx = threadIdx.x; idx < NCOL_TILES * 16 * F_DIM; idx += blockDim.x) {
      int r = idx >> 7, cc = idx & 127;
      smw[r * WLDS_STRIDE + cc] = (r < n_cols) ? W1[r * F_DIM + cc] : 0.0f;
    }
    __syncthreads();
    const float* __restrict__ arow = A1 + (size_t)ar * F_DIM;
    for (int k0 = 0; k0 < F_DIM; k0 += 4) {
      const int kk = k0 + krem;
      v2f a = *(const v2f*)(arow + kk);
#pragma unroll
      for (int t = 0; t < NCOL_TILES; ++t) {
        v2f b = *(const v2f*)(smw + (t * 16 + m) * WLDS_STRIDE + kk);
        c[t] = __builtin_amdgcn_wmma_f32_16x16x4_f32(
            false, a, false, b, (short)0, c[t], false, false);
      }
    }
  }

  // ---- epilogue: bias (+ReLU) and guarded stores ----
#pragma unroll
  for (int t = 0; t < NCOL_TILES; ++t) {
    const int n = t * 16 + m;
    if (n >= n_cols) continue;
    const float bv = bias[n];
#pragma unroll
    for (int v = 0; v < 8; ++v) {
      const int r = row0 + v + 8 * hi;
      if (r < n_rows) {
        float val = c[t][v] + bv;
        if (RELU) val = fmaxf(val, 0.0f);
        out[(size_t)r * n_cols + n] = val;
      }
    }
  }
}

// ---------------------------------------------------------------------------
// Orchestration
// ---------------------------------------------------------------------------
extern "C" void kernel_launch(void* const* d_in, const int* in_sizes, int n_in,
                              void* d_out, int out_size, void* d_ws, size_t ws_size,
                              hipStream_t stream) {
  // setup_inputs() order:
  const float* x        = (const float*)d_in[0];
  const int*   edge     = (const int*)d_in[1];   // [2, N_EDGES]
  const float* w1_neigh = (const float*)d_in[2];
  const float* w1_root  = (const float*)d_in[3];
  const float* b1       = (const float*)d_in[4];
  const float* w2_neigh = (const float*)d_in[5];
  const float* w2_root  = (const float*)d_in[6];
  const float* b2       = (const float*)d_in[7];
  const float* fc1_w    = (const float*)d_in[8];
  const float* fc1_b    = (const float*)d_in[9];
  const float* fc2_w    = (const float*)d_in[10];
  const float* fc2_b    = (const float*)d_in[11];
  float* out = (float*)d_out;

  const int* src = edge;
  const int* dst = edge + N_EDGES;

  const size_t NF = (size_t)N_NODES * F_DIM;  // 6.4M floats = 25.6 MB
  float* AGG = (float*)d_ws;      // reused for both layers' aggregation
  float* H1  = AGG + NF;          // layer1 output, later reused for fc1 out
  float* H2  = H1 + NF;           // layer2 output

  const int ZB = 256, ZG = (int)((NF + ZB - 1) / ZB);
  const int SB = 256, SG = (N_EDGES * 32 + SB - 1) / SB;

  // GEMM launch geometry: 4 waves/block, one 16-row tile per wave.
  const int row_tiles = (N_NODES + 15) / 16;       // 3125 (exact)
  const int GB = 128;                              // 4 waves
  const int GG = (row_tiles + 3) / 4;              // 782 blocks
  const size_t LDS8 = (size_t)(8 * 16) * WLDS_STRIDE * sizeof(float);  // 67584 B
  const size_t LDS1 = (size_t)(1 * 16) * WLDS_STRIDE * sizeof(float);  //  8448 B

  // Layer 1: agg = scatter(x); H1 = relu(x@w1_root^T + agg@w1_neigh^T + b1)
  zero_f32<<<ZG, ZB, 0, stream>>>(AGG, (int)NF);
  scatter_add_128<<<SG, SB, 0, stream>>>(x, src, dst, AGG, N_EDGES);
  gemm_wmma_kernel<8, true, true><<<GG, GB, LDS8, stream>>>(
      x, w1_root, AGG, w1_neigh, b1, H1, N_NODES, F_DIM);

  // Layer 2: agg = scatter(H1); H2 = relu(H1@w2_root^T + agg@w2_neigh^T + b2)
  zero_f32<<<ZG, ZB, 0, stream>>>(AGG, (int)NF);
  scatter_add_128<<<SG, SB, 0, stream>>>(H1, src, dst, AGG, N_EDGES);
  gemm_wmma_kernel<8, true, true><<<GG, GB, LDS8, stream>>>(
      H1, w2_root, AGG, w2_neigh, b2, H2, N_NODES, F_DIM);

  // fc1: H1 = relu(H2 @ fc1_w^T + fc1_b)   (H1 buffer reused)
  gemm_wmma_kernel<8, true, false><<<GG, GB, LDS8, stream>>>(
      H2, fc1_w, nullptr, nullptr, fc1_b, H1, N_NODES, F_DIM);

  // fc2: out = H1 @ fc2_w^T + fc2_b  -> [N_NODES, 10]
  gemm_wmma_kernel<1, false, false><<<GG, GB, LDS1, stream>>>(
      H1, fc2_w, nullptr, nullptr, fc2_b, out, N_NODES, N_CLASSES);
}